// MP_preprocessing_34686155882689
// MI455X (gfx1250) — compile-verified
//
#include <hip/hip_runtime.h>
#include <hip/hip_bf16.h>
#include <stdint.h>

#define BLOCK 256
#define EPT   8
#define TILE  (BLOCK * EPT)   // 2048 edges per tile
#define F_EDGE 8              // 8 f32 per edge row = 32 bytes

// Workgroup-relative LDS byte offset: on AMDGPU the generic pointer to LDS is
// {aperture_hi32, lds_offset32}, so the low 32 bits are the DS-usable offset.
__device__ __forceinline__ uint32_t lds_off_u32(const void* p) {
  return (uint32_t)(uintptr_t)p;
}

// gfx1250 async copy engine: copies 4 bytes at global sbase + g_byte_off to the LDS offset lds_byte_off (B32, GVS mode).
// Tracked by ASYNCcnt, not LOADcnt. NT hint: single-use stream, don't let it
// evict the hot atomic accumulator lines from L2.
__device__ __forceinline__ void async_gather_b32(uint32_t lds_byte_off,
                                                 uint32_t g_byte_off,
                                                 const void* sbase) {
  asm volatile("global_load_async_to_lds_b32 %0, %1, %2 th:TH_LOAD_NT"
               :
               : "v"(lds_byte_off), "v"(g_byte_off), "s"(sbase)
               : "memory");
}

__device__ __forceinline__ void wait_async_le8() {
  asm volatile("s_wait_asynccnt 0x8" ::: "memory");  // == EPT outstanding
}
__device__ __forceinline__ void wait_async_0() {
  asm volatile("s_wait_asynccnt 0x0" ::: "memory");
}

__global__ void __launch_bounds__(BLOCK)
seg_zero_kernel(float* __restrict__ sums, float* __restrict__ cnts, int n) {
  int i = blockIdx.x * BLOCK + threadIdx.x;
  if (i < n) { sums[i] = 0.0f; cnts[i] = 0.0f; }
}

__global__ void __launch_bounds__(BLOCK)
seg_scatter_kernel(const float* __restrict__ edge_x,
                   const int*   __restrict__ dst,
                   float* __restrict__ sums,
                   float* __restrict__ cnts,
                   int n_edges) {
  __shared__ float stage[2][TILE];   // 16 KB double buffer (of 320 KB WGP LDS)
  const int tid   = threadIdx.x;
  const int nfull = n_edges / TILE;

  int cur = blockIdx.x;
  int buf = 0;

  // Prologue: stage first tile's column-0 gather through the async engine.
  if (cur < nfull) {
    long long ebase = (long long)cur * TILE;
#pragma unroll
    for (int k = 0; k < EPT; ++k) {
      int slot = k * BLOCK + tid;                          // interleaved lanes -> dense line use
      uint32_t goff = (uint32_t)((ebase + slot) * (F_EDGE * 4));
      async_gather_b32(lds_off_u32(&stage[0][slot]), goff, (const void*)edge_x);
    }
  }

  while (cur < nfull) {
    int nxt = cur + (int)gridDim.x;
    if (nxt < nfull) {
      // Software pipeline: issue next tile into the other buffer first,
      // then wait until only those EPT newest loads are outstanding
      // (async loads complete in order -> current tile's data is in LDS).
      long long ebase = (long long)nxt * TILE;
#pragma unroll
      for (int k = 0; k < EPT; ++k) {
        int slot = k * BLOCK + tid;
        uint32_t goff = (uint32_t)((ebase + slot) * (F_EDGE * 4));
        async_gather_b32(lds_off_u32(&stage[buf ^ 1][slot]), goff, (const void*)edge_x);
      }
      wait_async_le8();
    } else {
      wait_async_0();
    }

    long long ebase = (long long)cur * TILE;
#pragma unroll
    for (int k = 0; k < EPT; ++k) {
      int slot = k * BLOCK + tid;
      float m = stage[buf][slot];                          // ds_load_b32, bank-conflict-free
      int d = __builtin_nontemporal_load(&dst[ebase + slot]); // coalesced NT stream
      atomicAdd(&sums[d], m);                              // global_atomic_add_f32 (L2-resident bins)
      atomicAdd(&cnts[d], 1.0f);
    }
    cur = nxt;
    buf ^= 1;
  }

  // Tail (< TILE edges): plain guarded loads — async loads must not run OOB.
  if (blockIdx.x == 0) {
    long long ebase = (long long)nfull * TILE;
#pragma unroll
    for (int k = 0; k < EPT; ++k) {
      long long e = ebase + (long long)k * BLOCK + tid;
      if (e < (long long)n_edges) {
        float m = __builtin_nontemporal_load(&edge_x[e * F_EDGE]);
        int d = __builtin_nontemporal_load(&dst[e]);
        atomicAdd(&sums[d], m);
        atomicAdd(&cnts[d], 1.0f);
      }
    }
  }
}

__global__ void __launch_bounds__(BLOCK)
seg_finalize_kernel(float* __restrict__ out,            // holds sums in-place
                    const float* __restrict__ cnts, int n) {
  int i = blockIdx.x * BLOCK + threadIdx.x;
  if (i < n) {
    float s = out[i];
    float c = cnts[i];
    out[i] = (c > 0.0f) ? (s / c) : 0.0f;
  }
}

extern "C" void kernel_launch(void* const* d_in, const int* in_sizes, int n_in,
                              void* d_out, int out_size, void* d_ws, size_t ws_size,
                              hipStream_t stream) {
  (void)n_in; (void)ws_size;
  const float* edge_x = (const float*)d_in[0];
  const int*   dst    = (const int*)d_in[1];   // harness delivers integer inputs as int32
  const int n_edges = in_sizes[1];
  const int n_nodes = out_size;                // == num_nodes

  float* sums = (float*)d_out;                 // accumulate sums directly into output
  float* cnts = (float*)d_ws;                  // 400 KB degree counters in scratch

  const int zgrid = (n_nodes + BLOCK - 1) / BLOCK;
  seg_zero_kernel<<<zgrid, BLOCK, 0, stream>>>(sums, cnts, n_nodes);

  const int nfull  = n_edges / TILE;
  int blocks = nfull < 512 ? (nfull > 0 ? nfull : 1) : 512;
  seg_scatter_kernel<<<blocks, BLOCK, 0, stream>>>(edge_x, dst, sums, cnts, n_edges);

  seg_finalize_kernel<<<zgrid, BLOCK, 0, stream>>>(sums, cnts, n_nodes);
}